// GraphTransformerFeatureExtractor_6219112644670
// MI455X (gfx1250) — compile-verified
//
#include <hip/hip_runtime.h>

#define NEG_SLOPE 0.2f

typedef float v2f __attribute__((ext_vector_type(2)));
typedef float v8f __attribute__((ext_vector_type(8)));

// ---------- helpers: order-preserving float<->uint for atomicMax ----------
__device__ __forceinline__ unsigned fenc(float f) {
    unsigned u = __float_as_uint(f);
    return (u & 0x80000000u) ? ~u : (u | 0x80000000u);
}
__device__ __forceinline__ float fdec(unsigned k) {
    unsigned u = (k & 0x80000000u) ? (k ^ 0x80000000u) : ~k;
    return __uint_as_float(u);
}

// ---------- 1) projection: [N,7] @ [128,7]^T + b -> [N,128] ----------
__global__ void proj_kernel(const float* __restrict__ x, const float* __restrict__ w,
                            const float* __restrict__ b, float* __restrict__ out, int nNodes) {
    int t = blockIdx.x * blockDim.x + threadIdx.x;
    if (t >= nNodes * 128) return;
    int n = t >> 7, d = t & 127;
    const float* xr = x + (size_t)n * 7;
    const float* wr = w + (size_t)d * 7;
    float s = b[d];
#pragma unroll
    for (int i = 0; i < 7; ++i) s += xr[i] * wr[i];
    out[t] = s;
}

// ---------- 2) WMMA fp32 GEMM: Out[n][o] = sum_k A[n][k] * W[o][k] (+bias, relu) ----------
// K fixed = 128. Block = 32 * (nCols/16) threads; one 16x16 tile per wave.
__global__ void gemm_wmma_f32(const float* __restrict__ A, const float* __restrict__ W,
                              const float* __restrict__ bias, float* __restrict__ Out,
                              int nRows, int nCols, int relu) {
    __shared__ float As[16 * 128];
    const int tid = threadIdx.x;
    const int rowBase = blockIdx.x * 16;
    // cooperative, coalesced load of the 16x128 A tile (rows are contiguous)
    {
        const float* src = A + (size_t)rowBase * 128;
        const int limit = (nRows - rowBase) * 128;   // guard last tile
        for (int i = tid; i < 16 * 128; i += blockDim.x)
            As[i] = (i < limit) ? src[i] : 0.0f;
    }
    __syncthreads();

    const int wave = tid >> 5;
    const int lane = tid & 31;
    const int col = wave * 16 + (lane & 15);   // output column (N dim)
    const int row = lane & 15;                 // A row within tile (M dim)
    const int kg  = (lane >> 4) * 2;           // K sub-pair select per ISA layout

    v8f acc = {0.f, 0.f, 0.f, 0.f, 0.f, 0.f, 0.f, 0.f};
    const float* wrow = W + (size_t)col * 128; // B[k][col] = W[col][k]
#pragma unroll 4
    for (int k0 = 0; k0 < 128; k0 += 4) {
        v2f a, b;
        a.x = As[row * 128 + k0 + kg];
        a.y = As[row * 128 + k0 + kg + 1];
        b.x = wrow[k0 + kg];
        b.y = wrow[k0 + kg + 1];
        acc = __builtin_amdgcn_wmma_f32_16x16x4_f32(
            /*neg_a=*/false, a, /*neg_b=*/false, b,
            /*c_mod=*/(short)0, acc, /*reuse_a=*/false, /*reuse_b=*/false);
    }

    const int r0 = rowBase + ((lane < 16) ? 0 : 8);
    const float bv = bias ? bias[col] : 0.0f;
#pragma unroll
    for (int v = 0; v < 8; ++v) {
        int r = r0 + v;
        if (r < nRows) {
            float val = acc[v] + bv;
            if (relu) val = fmaxf(val, 0.0f);
            Out[(size_t)r * nCols + col] = val;
        }
    }
}

// ---------- 3) attention coefficients alpha_s/alpha_d : [N,8] ----------
__global__ void alpha_kernel(const float* __restrict__ hw, const float* __restrict__ asrc,
                             const float* __restrict__ adst, float* __restrict__ als,
                             float* __restrict__ ald, int nNodes) {
    int t = blockIdx.x * blockDim.x + threadIdx.x;
    if (t >= nNodes * 8) return;
    int n = t >> 3, hh = t & 7;
    const float* hr = hw + (size_t)n * 128 + hh * 16;
    float s = 0.f, d = 0.f;
#pragma unroll
    for (int c = 0; c < 16; ++c) {
        s += hr[c] * asrc[hh * 16 + c];
        d += hr[c] * adst[hh * 16 + c];
    }
    als[t] = s;
    ald[t] = d;
}

// ---------- 4a) edge logits + segment max (monotone-uint atomicMax) ----------
__global__ void edge_logits_max(const int* __restrict__ ei, const float* __restrict__ als,
                                const float* __restrict__ ald, float* __restrict__ elog,
                                unsigned* __restrict__ mkey, int E, int Ep) {
    int t = blockIdx.x * blockDim.x + threadIdx.x;
    if (t >= Ep * 8) return;
    int e = t >> 3, hh = t & 7;
    int s, d;
    if (e < E) { s = ei[e]; d = ei[E + e]; } else { s = d = e - E; }
    float v = als[s * 8 + hh] + ald[d * 8 + hh];
    v = (v >= 0.f) ? v : NEG_SLOPE * v;
    elog[t] = v;
    atomicMax(&mkey[d * 8 + hh], fenc(v));
}

// ---------- 4b) exp(e - max) + segment sum ----------
__global__ void edge_exp_sum(const int* __restrict__ ei, float* __restrict__ elog,
                             const unsigned* __restrict__ mkey, float* __restrict__ denom,
                             int E, int Ep) {
    int t = blockIdx.x * blockDim.x + threadIdx.x;
    if (t >= Ep * 8) return;
    int e = t >> 3, hh = t & 7;
    int d = (e < E) ? ei[E + e] : (e - E);
    float m = fdec(mkey[d * 8 + hh]);
    float ex = __expf(elog[t] - m);
    elog[t] = ex;
    atomicAdd(&denom[d * 8 + hh], ex);
}

// ---------- 4c) weighted scatter-add of source messages ----------
__global__ void edge_scatter(const int* __restrict__ ei, const float* __restrict__ hw,
                             const float* __restrict__ elog, const float* __restrict__ denom,
                             float* __restrict__ acc, int E, int Ep) {
    int t = blockIdx.x * blockDim.x + threadIdx.x;
    if (t >= Ep * 8) return;
    int e = t >> 3, hh = t & 7;
    int s, d;
    if (e < E) { s = ei[e]; d = ei[E + e]; } else { s = d = e - E; }
    float coef = elog[t] / denom[d * 8 + hh];
    const float4* hr = reinterpret_cast<const float4*>(hw + (size_t)s * 128 + hh * 16);
    float* ar = acc + (size_t)d * 128 + hh * 16;
#pragma unroll
    for (int q = 0; q < 4; ++q) {
        float4 hv = hr[q];
        atomicAdd(ar + q * 4 + 0, hv.x * coef);
        atomicAdd(ar + q * 4 + 1, hv.y * coef);
        atomicAdd(ar + q * 4 + 2, hv.z * coef);
        atomicAdd(ar + q * 4 + 3, hv.w * coef);
    }
}

// ---------- 5) finalize GAT layer: relu(acc + bias) ----------
__global__ void bias_relu_kernel(const float* __restrict__ acc, const float* __restrict__ b,
                                 float* __restrict__ out, int nNodes) {
    int t = blockIdx.x * blockDim.x + threadIdx.x;
    if (t >= nNodes * 128) return;
    out[t] = fmaxf(acc[t] + b[t & 127], 0.0f);
}

// ---------- 6) fe2: [N,64] @ [7,64]^T + b -> [N,7] ----------
__global__ void fe2_kernel(const float* __restrict__ h, const float* __restrict__ w,
                           const float* __restrict__ b, float* __restrict__ out, int nNodes) {
    int t = blockIdx.x * blockDim.x + threadIdx.x;
    if (t >= nNodes * 7) return;
    int n = t / 7, j = t % 7;
    const float* hr = h + (size_t)n * 64;
    const float* wr = w + (size_t)j * 64;
    float s = b[j];
#pragma unroll 8
    for (int i = 0; i < 64; ++i) s += hr[i] * wr[i];
    out[t] = s;
}

extern "C" void kernel_launch(void* const* d_in, const int* in_sizes, int n_in,
                              void* d_out, int out_size, void* d_ws, size_t ws_size,
                              hipStream_t stream) {
    const float* x      = (const float*)d_in[0];
    const int*   ei     = (const int*)d_in[1];
    const float* proj_w = (const float*)d_in[2];
    const float* proj_b = (const float*)d_in[3];
    const float* gw[2]  = {(const float*)d_in[4], (const float*)d_in[8]};
    const float* gas[2] = {(const float*)d_in[5], (const float*)d_in[9]};
    const float* gad[2] = {(const float*)d_in[6], (const float*)d_in[10]};
    const float* gb[2]  = {(const float*)d_in[7], (const float*)d_in[11]};
    const float* fe1_w  = (const float*)d_in[12];
    const float* fe1_b  = (const float*)d_in[13];
    const float* fe2_w  = (const float*)d_in[14];
    const float* fe2_b  = (const float*)d_in[15];

    const int N  = in_sizes[0] / 7;   // 50000
    const int E  = in_sizes[1] / 2;   // 800000
    const int Ep = E + N;             // + self loops

    // scratch layout
    char* p = (char*)d_ws;
    float* bufH = (float*)p; p += (size_t)N * 128 * sizeof(float); // layer input h
    float* bufW = (float*)p; p += (size_t)N * 128 * sizeof(float); // h @ W^T
    float* bufA = (float*)p; p += (size_t)N * 128 * sizeof(float); // scatter accumulator
    float* als  = (float*)p; p += (size_t)N * 8 * sizeof(float);
    float* ald  = (float*)p; p += (size_t)N * 8 * sizeof(float);
    unsigned* mkey = (unsigned*)p; p += (size_t)N * 8 * sizeof(unsigned);
    float* denom = (float*)p; p += (size_t)N * 8 * sizeof(float);
    float* elog  = (float*)p; p += (size_t)Ep * 8 * sizeof(float);

    auto cdiv = [](long long a, long long b) { return (int)((a + b - 1) / b); };

    proj_kernel<<<cdiv((long long)N * 128, 256), 256, 0, stream>>>(x, proj_w, proj_b, bufH, N);

    for (int L = 0; L < 2; ++L) {
        gemm_wmma_f32<<<cdiv(N, 16), 256, 0, stream>>>(bufH, gw[L], nullptr, bufW, N, 128, 0);
        alpha_kernel<<<cdiv((long long)N * 8, 256), 256, 0, stream>>>(bufW, gas[L], gad[L], als, ald, N);
        hipMemsetAsync(mkey, 0, (size_t)N * 8 * sizeof(unsigned), stream);
        hipMemsetAsync(denom, 0, (size_t)N * 8 * sizeof(float), stream);
        hipMemsetAsync(bufA, 0, (size_t)N * 128 * sizeof(float), stream);
        edge_logits_max<<<cdiv((long long)Ep * 8, 256), 256, 0, stream>>>(ei, als, ald, elog, mkey, E, Ep);
        edge_exp_sum<<<cdiv((long long)Ep * 8, 256), 256, 0, stream>>>(ei, elog, mkey, denom, E, Ep);
        edge_scatter<<<cdiv((long long)Ep * 8, 256), 256, 0, stream>>>(ei, bufW, elog, denom, bufA, E, Ep);
        bias_relu_kernel<<<cdiv((long long)N * 128, 256), 256, 0, stream>>>(bufA, gb[L], bufH, N);
    }

    // fe1: [N,128] -> [N,64], fused bias + relu (4 waves / block)
    gemm_wmma_f32<<<cdiv(N, 16), 128, 0, stream>>>(bufH, fe1_w, fe1_b, bufW, N, 64, 1);
    // fe2: [N,64] -> [N,7]
    fe2_kernel<<<cdiv((long long)N * 7, 256), 256, 0, stream>>>(bufW, fe2_w, fe2_b, (float*)d_out, N);
}